// BaseCostVolume3D_48060684043140
// MI455X (gfx1250) — compile-verified
//
#include <hip/hip_runtime.h>

// Problem dims (fixed by the reference).
#define N_  2
#define C_  32
#define D_  48
#define H_  136
#define W_  240
#define PAD 48   // zero pad in front of right row so (w - d) < 0 reads zero

typedef float f32x4 __attribute__((ext_vector_type(4)));
typedef int   v4i   __attribute__((ext_vector_type(4)));
typedef __attribute__((address_space(1))) v4i* g4ptr;  // global int4*
typedef __attribute__((address_space(3))) v4i* l4ptr;  // LDS int4*

__global__ __launch_bounds__(256)
void cost_volume_kernel(const float* __restrict__ left,
                        const float* __restrict__ right,
                        float* __restrict__ out) {
    __shared__ float sL[W_];         // left row
    __shared__ float sR[PAD + W_];   // [0,48) = zeros, [48,288) = right row

    const int h   = blockIdx.x;   // 0..135
    const int nc  = blockIdx.y;   // 0..63  (n*C + c)
    const int tid = threadIdx.x;  // 0..255

    const float* lrow = left  + ((size_t)nc * H_ + h) * W_;
    const float* rrow = right + ((size_t)nc * H_ + h) * W_;

    // ---- Stage the two input rows into LDS via gfx1250 async-to-LDS DMA ----
    // 60 x b128 per row; global rows start at 960B multiples, LDS dsts 16B aligned.
    if (tid < 60) {
        __builtin_amdgcn_global_load_async_to_lds_b128(
            (g4ptr)(v4i*)(lrow + tid * 4),
            (l4ptr)(v4i*)(&sL[tid * 4]), 0, 0);
    } else if (tid >= 64 && tid < 124) {
        const int t = tid - 64;
        __builtin_amdgcn_global_load_async_to_lds_b128(
            (g4ptr)(v4i*)(rrow + t * 4),
            (l4ptr)(v4i*)(&sR[PAD + t * 4]), 0, 0);
    } else if (tid >= 128 && tid < 128 + PAD) {
        sR[tid - 128] = 0.0f;        // branch-free left boundary
    }
    asm volatile("s_wait_asynccnt 0" ::: "memory");
    __syncthreads();

    // ---- Fan out: each thread owns one float4 column and one d-subgroup ----
    const int w4 = tid & 63;   // float4 column 0..63 (60 valid)
    const int dg = tid >> 6;   // d subgroup 0..3
    if (w4 >= 60) return;

    const int    wb = w4 * 4;
    const f32x4  l  = *(const f32x4*)(&sL[wb]);   // loop-invariant left vec
    const float* rp = &sR[PAD];                   // rp[i] valid for i in [-48, 240)

    float* obase = out + ((size_t)nc * D_ * H_ + h) * W_ + wb;

    #pragma unroll
    for (int i = 0; i < 12; ++i) {
        const int d = dg + 4 * i;                 // covers d = 0..47
        f32x4 v;
        v.x = l.x * rp[wb + 0 - d];
        v.y = l.y * rp[wb + 1 - d];
        v.z = l.z * rp[wb + 2 - d];
        v.w = l.w * rp[wb + 3 - d];
        // Output is write-once, > L2 capacity: non-temporal b128 store.
        __builtin_nontemporal_store(v, (f32x4*)(obase + (size_t)d * (H_ * W_)));
    }
}

extern "C" void kernel_launch(void* const* d_in, const int* in_sizes, int n_in,
                              void* d_out, int out_size, void* d_ws, size_t ws_size,
                              hipStream_t stream) {
    const float* left  = (const float*)d_in[0];
    const float* right = (const float*)d_in[1];
    float*       out   = (float*)d_out;

    dim3 grid(H_, N_ * C_);   // one block per (n, c, h) row: 136 x 64 = 8704
    cost_volume_kernel<<<grid, 256, 0, stream>>>(left, right, out);
}